// RWKV_LAYER_40699110097200
// MI455X (gfx1250) — compile-verified
//
#include <hip/hip_runtime.h>
#include <hip/hip_bf16.h>

typedef __attribute__((ext_vector_type(2))) float v2f;
typedef __attribute__((ext_vector_type(8))) float v8f;

#define LYR 12
#define DM  2048
#define DF  8192

// ---------------------------------------------------------------------------
// WMMA-based partial dot: 16 rows of W (row-major, leading dim ld) against
// vector xin, over NCOL columns starting at c0.  V_WMMA_F32_16X16X4_F32 with
// A = 16x4 W tile (lane m -> row rb+m, contiguous float2 = b64 load) and
// B = x broadcast across all 16 N columns.  Result: lane 0 holds y[rb+0..7]
// in acc[0..7], lane 16 holds y[rb+8..15].  NCOL is compile-time so the loop
// uses a scalar counter (no per-lane exec masking); two accumulators break
// the WMMA->WMMA RAW chain.
// ---------------------------------------------------------------------------
template <int NCOL>
__device__ __forceinline__ v8f wave_dot16(const float* __restrict__ W, int ld, int rb,
                                          const float* __restrict__ xin, int c0) {
  const int lane = threadIdx.x & 31;
  const int m    = lane & 15;
  const int kOff = (lane >> 4) << 1;   // lanes 0-15: K=0,1   lanes 16-31: K=2,3
  const float* wr = W + (size_t)(rb + m) * (size_t)ld + (c0 + kOff);
  const float* xp = xin + (c0 + kOff);
  v8f acc0 = {}, acc1 = {};
#pragma unroll 4
  for (int t = 0; t < NCOL; t += 8) {
    v2f a0 = *(const v2f*)(wr + t);
    v2f b0 = *(const v2f*)(xp + t);
    v2f a1 = *(const v2f*)(wr + t + 4);
    v2f b1 = *(const v2f*)(xp + t + 4);
    acc0 = __builtin_amdgcn_wmma_f32_16x16x4_f32(false, a0, false, b0, (short)0, acc0,
                                                 false, false);
    acc1 = __builtin_amdgcn_wmma_f32_16x16x4_f32(false, a1, false, b1, (short)0, acc1,
                                                 false, false);
  }
  return acc0 + acc1;
}

// Lane 0 / lane 16 deposit their 8 row-partials into LDS for cross-wave reduce.
__device__ __forceinline__ void deposit16(float (*red)[16], v8f acc) {
  const int lane = threadIdx.x & 31;
  const int wv   = threadIdx.x >> 5;
  if ((lane & 15) == 0) {
    const int half = (lane >> 4) << 3;
#pragma unroll
    for (int i = 0; i < 8; ++i) red[wv][half + i] = acc[i];
  }
}

__device__ __forceinline__ float sum8(const float (*red)[16], int m) {
  float s = 0.f;
#pragma unroll
  for (int w = 0; w < 8; ++w) s += red[w][m];
  return s;
}

// ---------------------------------------------------------------------------
// LayerNorm + token-shift mix.  Single block of 256 threads, 8 elems/thread.
// xln = (x-mu)*rsqrt(var+eps)*w + b ; o_j = xln + mixw_j * state_row
// ---------------------------------------------------------------------------
__global__ void ln_mix_kernel(const float* __restrict__ xin,
                              const float* __restrict__ w, const float* __restrict__ b,
                              const float* __restrict__ mix0,
                              const float* __restrict__ mix1,
                              const float* __restrict__ mix2,
                              const float* __restrict__ srow,
                              float* __restrict__ xln_out,
                              float* __restrict__ o0, float* __restrict__ o1,
                              float* __restrict__ o2) {
  __shared__ float sred[256];
  const int t = threadIdx.x;
  float vals[8];
  float s = 0.f;
#pragma unroll
  for (int i = 0; i < 8; ++i) { vals[i] = xin[t + 256 * i]; s += vals[i]; }
  sred[t] = s; __syncthreads();
  for (int st = 128; st > 0; st >>= 1) { if (t < st) sred[t] += sred[t + st]; __syncthreads(); }
  const float mu = sred[0] * (1.f / (float)DM);
  __syncthreads();
  float s2 = 0.f;
#pragma unroll
  for (int i = 0; i < 8; ++i) { float d = vals[i] - mu; s2 += d * d; }
  sred[t] = s2; __syncthreads();
  for (int st = 128; st > 0; st >>= 1) { if (t < st) sred[t] += sred[t + st]; __syncthreads(); }
  const float rs = rsqrtf(sred[0] * (1.f / (float)DM) + 1e-5f);
#pragma unroll
  for (int i = 0; i < 8; ++i) {
    const int idx = t + 256 * i;
    const float xl = (vals[i] - mu) * rs * w[idx] + b[idx];
    xln_out[idx] = xl;
    const float sa = srow[idx];
    o0[idx] = xl + mix0[idx] * sa;
    o1[idx] = xl + mix1[idx] * sa;
    if (o2) o2[idx] = xl + mix2[idx] * sa;
  }
}

// ---------------------------------------------------------------------------
// Fused K/V/R mat-vecs + WKV combine + state update.
// One block = one 16-row tile across all three matrices; 8 waves split columns.
//   k = exp(K@xk), v = V@xv, rr = exp((-R)@xr)
//   wkv = (aa + v*k*tf) / ((rr+1)*(k*tf + bb))
//   nb  = aa*td + k*v ;  nc = bb*td + k
// ---------------------------------------------------------------------------
__global__ void kvr_kernel(const float* __restrict__ Wkvr,
                           const float* __restrict__ xk, const float* __restrict__ xv,
                           const float* __restrict__ xr,
                           const float* __restrict__ aa_row, const float* __restrict__ bb_row,
                           const float* __restrict__ tf_row, const float* __restrict__ td_row,
                           float* __restrict__ wkv,
                           float* __restrict__ nb, float* __restrict__ nc) {
  __shared__ float red[3][8][16];
  const int rb = blockIdx.x * 16;
  const int wv = threadIdx.x >> 5;
  const int c0 = wv * (DM / 8);
  const float* WK = Wkvr;
  const float* WV = Wkvr + (size_t)DM * DM;
  const float* WR = Wkvr + 2 * (size_t)DM * DM;
  v8f ak = wave_dot16<DM / 8>(WK, DM, rb, xk, c0);
  v8f av = wave_dot16<DM / 8>(WV, DM, rb, xv, c0);
  v8f ar = wave_dot16<DM / 8>(WR, DM, rb, xr, c0);
  deposit16(red[0], ak);
  deposit16(red[1], av);
  deposit16(red[2], ar);
  __syncthreads();
  if (threadIdx.x < 16) {
    const int m = threadIdx.x, i = rb + m;
    const float kk = __expf(sum8(red[0], m));
    const float vv = sum8(red[1], m);
    const float rr = __expf(sum8(red[2], m));
    const float aa = aa_row[i], bb = bb_row[i];
    const float tf = tf_row[i], td = td_row[i];
    const float num = aa + vv * kk * tf;
    const float den = (rr + 1.f) * (kk * tf + bb);
    wkv[i] = num / den;
    nb[i]  = aa * td + kk * vv;
    nc[i]  = bb * td + kk;
  }
}

// sx = addend + W @ x   (output projection)
__global__ void gemv_add_kernel(const float* __restrict__ W, const float* __restrict__ xin,
                                const float* __restrict__ addend, float* __restrict__ y) {
  __shared__ float red[8][16];
  const int rb = blockIdx.x * 16;
  const int wv = threadIdx.x >> 5;
  v8f a = wave_dot16<DM / 8>(W, DM, rb, xin, wv * (DM / 8));
  deposit16(red, a);
  __syncthreads();
  if (threadIdx.x < 16) {
    const int i = rb + threadIdx.x;
    y[i] = addend[i] + sum8(red, threadIdx.x);
  }
}

// kf = relu(W @ x)^2   (FFN key, 8192 rows over 2048 cols)
__global__ void gemv_sqrelu_kernel(const float* __restrict__ W, const float* __restrict__ xin,
                                   float* __restrict__ y) {
  __shared__ float red[8][16];
  const int rb = blockIdx.x * 16;
  const int wv = threadIdx.x >> 5;
  v8f a = wave_dot16<DM / 8>(W, DM, rb, xin, wv * (DM / 8));
  deposit16(red, a);
  __syncthreads();
  if (threadIdx.x < 16) {
    const int i = rb + threadIdx.x;
    float s = sum8(red, threadIdx.x);
    s = fmaxf(s, 0.f);
    y[i] = s * s;
  }
}

// out = sx + sigmoid(RF@xr2) * (VF@kf)   (fused FFN receptance + value + gate)
__global__ void ffnout_kernel(const float* __restrict__ RF, const float* __restrict__ xr2,
                              const float* __restrict__ VF, const float* __restrict__ kf,
                              const float* __restrict__ sx, float* __restrict__ xout) {
  __shared__ float red[2][8][16];
  const int rb = blockIdx.x * 16;
  const int wv = threadIdx.x >> 5;
  v8f ar = wave_dot16<DM / 8>(RF, DM, rb, xr2, wv * (DM / 8));
  v8f ao = wave_dot16<DF / 8>(VF, DF, rb, kf,  wv * (DF / 8));
  deposit16(red[0], ar);
  deposit16(red[1], ao);
  __syncthreads();
  if (threadIdx.x < 16) {
    const int i = rb + threadIdx.x;
    const float r = 1.f / (1.f + __expf(-sum8(red[0], threadIdx.x)));
    xout[i] = sx[i] + r * sum8(red[1], threadIdx.x);
  }
}

__global__ void copy_kernel(const float* __restrict__ a, float* __restrict__ b) {
  const int i = blockIdx.x * blockDim.x + threadIdx.x;
  b[i] = a[i];
}

// ---------------------------------------------------------------------------
extern "C" void kernel_launch(void* const* d_in, const int* in_sizes, int n_in,
                              void* d_out, int out_size, void* d_ws, size_t ws_size,
                              hipStream_t stream) {
  (void)in_sizes; (void)n_in; (void)out_size; (void)ws_size;

  const float* x_in  = (const float*)d_in[0];
  const float* state = (const float*)d_in[1];
  const float* ln1w  = (const float*)d_in[2];
  const float* ln1b  = (const float*)d_in[3];
  const float* ln2w  = (const float*)d_in[4];
  const float* ln2b  = (const float*)d_in[5];
  const float* tdec  = (const float*)d_in[6];
  const float* tfir  = (const float*)d_in[7];
  const float* vvtv  = (const float*)d_in[8];
  const float* kktk  = (const float*)d_in[9];
  const float* rrtr  = (const float*)d_in[10];
  const float* keyw  = (const float*)d_in[11];   // (L,3,D,D)
  const float* outw  = (const float*)d_in[12];   // (L,D,D)
  const float* tmk   = (const float*)d_in[13];
  const float* tmr   = (const float*)d_in[14];
  const float* kffn  = (const float*)d_in[15];   // (L,H,D)
  const float* rffn  = (const float*)d_in[16];   // (L,D,D)
  const float* vffn  = (const float*)d_in[17];   // (L,D,H)

  float* out = (float*)d_out;
  float* na  = out + DM;            // xln per layer
  float* nb  = na + LYR * DM;       // new aa
  float* nc  = nb + LYR * DM;       // new bb
  float* nd  = nc + LYR * DM;       // x2 per layer

  float* ws   = (float*)d_ws;
  float* xcur = ws;                 // D   current residual stream
  float* xk   = xcur + DM;          // D
  float* xv   = xk + DM;            // D
  float* xr   = xv + DM;            // D
  float* wkv  = xr + DM;            // D
  float* sx   = wkv + DM;           // D
  float* xk2  = sx + DM;            // D
  float* xr2  = xk2 + DM;           // D
  float* kf   = xr2 + DM;           // H

  copy_kernel<<<DM / 256, 256, 0, stream>>>(x_in, xcur);

  for (int l = 0; l < LYR; ++l) {
    const size_t off = (size_t)l * DM;
    const float* statea = state + (size_t)(0 * LYR + l) * DM;
    const float* stateb = state + (size_t)(1 * LYR + l) * DM;
    const float* statec = state + (size_t)(2 * LYR + l) * DM;
    const float* stated = state + (size_t)(3 * LYR + l) * DM;

    // LN1 + token-shift mixes (against entry state)
    ln_mix_kernel<<<1, 256, 0, stream>>>(xcur, ln1w + off, ln1b + off,
                                         kktk + off, vvtv + off, rrtr + off,
                                         statea, na + off, xk, xv, xr);
    // K/V/R mat-vecs + WKV + state update
    kvr_kernel<<<DM / 16, 256, 0, stream>>>(keyw + (size_t)l * 3 * DM * DM,
                                            xk, xv, xr, stateb, statec,
                                            tfir + off, tdec + off,
                                            wkv, nb + off, nc + off);
    // sx = xcur + O @ wkv
    gemv_add_kernel<<<DM / 16, 256, 0, stream>>>(outw + (size_t)l * DM * DM,
                                                 wkv, xcur, sx);
    // LN2 + FFN token-shift mixes
    ln_mix_kernel<<<1, 256, 0, stream>>>(sx, ln2w + off, ln2b + off,
                                         tmk + off, tmr + off, nullptr,
                                         stated, nd + off, xk2, xr2, nullptr);
    // kf = relu(KF @ xk2)^2
    gemv_sqrelu_kernel<<<DF / 16, 256, 0, stream>>>(kffn + (size_t)l * DF * DM, xk2, kf);
    // xcur = sx + sigmoid(RF @ xr2) * (VF @ kf)
    ffnout_kernel<<<DM / 16, 256, 0, stream>>>(rffn + (size_t)l * DM * DM, xr2,
                                               vffn + (size_t)l * DM * DF, kf,
                                               sx, xcur);
  }

  copy_kernel<<<DM / 256, 256, 0, stream>>>(xcur, out);
}